// Centerdist_3547642986610
// MI455X (gfx1250) — compile-verified
//
#include <hip/hip_runtime.h>
#include <stdint.h>

// Center-distance loss, MI455X (gfx1250, wave32).
// Closed form per id: loss_k = (sum||x||^2 - ||sum x||^2 / n) / n
// Pass 1: stream 256MB once, atomically accumulate count / sum / sumsq per id.
// Pass 2: ||sum_k||^2 via v_wmma_f32_16x16x4_f32 Gram-diagonal per 16-id block.

typedef __attribute__((ext_vector_type(2))) float v2f;
typedef __attribute__((ext_vector_type(8))) float v8f;

#define NUM_IDS 1000
#define FEAT_D  256

// workspace layout (floats)
static constexpr int WS_SUMS   = 0;                    // [NUM_IDS * FEAT_D]
static constexpr int WS_COUNTS = NUM_IDS * FEAT_D;     // [NUM_IDS]
static constexpr int WS_SUMSQ  = WS_COUNTS + NUM_IDS;  // [NUM_IDS]
static constexpr int WS_LOSS   = WS_SUMSQ + NUM_IDS;   // scalar
static constexpr int WS_UNIQ   = WS_LOSS + 1;          // scalar
static constexpr int WS_HASNEG = WS_UNIQ + 1;          // scalar
static constexpr int WS_TOTAL  = WS_HASNEG + 1;        // 258003 floats ~= 1.01 MB

__global__ void cd_zero(float* __restrict__ ws) {
  int i = blockIdx.x * blockDim.x + threadIdx.x;
  if (i < WS_TOTAL) ws[i] = 0.0f;
}

// One wave (32 lanes) per row: lane handles 8 consecutive floats (two b128 loads).
__global__ __launch_bounds__(256)
void cd_accum(const float* __restrict__ feat, const int* __restrict__ ids,
              float* __restrict__ ws, int nrows) {
  float* sums   = ws + WS_SUMS;
  float* counts = ws + WS_COUNTS;
  float* sumsq  = ws + WS_SUMSQ;

  const int lane  = threadIdx.x & 31;
  const int wib   = threadIdx.x >> 5;
  const int wpb   = blockDim.x >> 5;
  const int gwave = blockIdx.x * wpb + wib;
  const int nwave = gridDim.x * wpb;

  for (int r = gwave; r < nrows; r += nwave) {
    const float* row = feat + (size_t)r * FEAT_D + lane * 8;
    if (r + nwave < nrows) {
      // gfx1250: lowers to global_prefetch_b8 — pull next tile toward L2/L0
      __builtin_prefetch(feat + (size_t)(r + nwave) * FEAT_D + lane * 8, 0, 3);
    }
    const float4 a = ((const float4*)row)[0];
    const float4 b = ((const float4*)row)[1];
    const int id = ids[r];

    float ss = a.x*a.x + a.y*a.y + a.z*a.z + a.w*a.w
             + b.x*b.x + b.y*b.y + b.z*b.z + b.w*b.w;
#pragma unroll
    for (int off = 16; off > 0; off >>= 1) ss += __shfl_xor(ss, off, 32);

    if (id >= 0) {
      float* s = sums + (size_t)id * FEAT_D + lane * 8;
      atomicAdd(s + 0, a.x); atomicAdd(s + 1, a.y);
      atomicAdd(s + 2, a.z); atomicAdd(s + 3, a.w);
      atomicAdd(s + 4, b.x); atomicAdd(s + 5, b.y);
      atomicAdd(s + 6, b.z); atomicAdd(s + 7, b.w);
      if (lane == 0) {
        atomicAdd(counts + id, 1.0f);
        atomicAdd(sumsq  + id, ss);
      }
    } else if (lane == 0) {
      ws[WS_HASNEG] = 1.0f;  // benign race: all writers store 1.0f
    }
  }
}

// One wave per 16-id block: diag(S S^T) via v_wmma_f32_16x16x4_f32, K=256 in steps of 4.
// For a Gram product the f32 A (16x4) and B (4x16) lane fragments coincide:
//   lane l, row/col m=l&15, lo-lanes carry K={k0,k0+1}, hi-lanes K={k0+2,k0+3}.
__global__ __launch_bounds__(32)
void cd_finalize(float* __restrict__ ws) {
  __shared__ float S[16 * FEAT_D];  // 16 KB
  const float* sums   = ws + WS_SUMS;
  const float* counts = ws + WS_COUNTS;
  const float* sumsq  = ws + WS_SUMSQ;

  const int lane = threadIdx.x;
  const int base = blockIdx.x * 16;

  // Stage the 16x256 sums block into LDS (zero-pad past NUM_IDS).
  for (int i = lane; i < 16 * FEAT_D / 4; i += 32) {
    const int m  = i >> 6;   // row 0..15
    const int c4 = i & 63;   // float4 within row
    const int id = base + m;
    float4 v = make_float4(0.f, 0.f, 0.f, 0.f);
    if (id < NUM_IDS) v = ((const float4*)(sums + (size_t)id * FEAT_D))[c4];
    ((float4*)S)[i] = v;
  }
  __syncthreads();

  const int  rowA = lane & 15;
  const bool hi   = lane >= 16;
  v8f c = {0.f, 0.f, 0.f, 0.f, 0.f, 0.f, 0.f, 0.f};
#pragma unroll 4
  for (int k0 = 0; k0 < FEAT_D; k0 += 4) {
    v2f f;
    f.x = S[rowA * FEAT_D + k0 + (hi ? 2 : 0)];
    f.y = S[rowA * FEAT_D + k0 + (hi ? 3 : 1)];
    // 8 args: (neg_a, A, neg_b, B, c_mod, C, reuse_a, reuse_b)
    c = __builtin_amdgcn_wmma_f32_16x16x4_f32(false, f, false, f, (short)0, c,
                                              false, false);
  }

  // Diagonal (m,m): acc VGPR m%8 at lane m (m<8) or lane m+16 (m>=8).
  int m = -1, idx = -1;
  if (lane < 8)        { m = lane;      idx = lane;      }
  else if (lane >= 24) { m = lane - 16; idx = lane - 24; }
  float diag = 0.f;
  if (idx >= 0) {
#pragma unroll
    for (int i = 0; i < 8; ++i) if (idx == i) diag = c[i];
  }

  float contrib = 0.f, uniq = 0.f;
  if (m >= 0) {
    const int id = base + m;
    if (id < NUM_IDS) {
      const float n = counts[id];
      if (n > 0.f) {
        contrib = (sumsq[id] - diag / n) / n;  // per-id loss
        uniq = 1.f;
      }
    }
  }
#pragma unroll
  for (int off = 16; off > 0; off >>= 1) {
    contrib += __shfl_xor(contrib, off, 32);
    uniq    += __shfl_xor(uniq,    off, 32);
  }
  if (lane == 0) {
    atomicAdd(ws + WS_LOSS, contrib);
    atomicAdd(ws + WS_UNIQ, uniq);
  }
}

__global__ void cd_final(const float* __restrict__ ws, float* __restrict__ out) {
  float denom = ws[WS_UNIQ] + ws[WS_HASNEG];
  denom = denom > 0.f ? denom : 1.f;
  out[0] = ws[WS_LOSS] / denom;
}

extern "C" void kernel_launch(void* const* d_in, const int* in_sizes, int n_in,
                              void* d_out, int out_size, void* d_ws, size_t ws_size,
                              hipStream_t stream) {
  (void)n_in; (void)out_size; (void)ws_size;
  const float* feat = (const float*)d_in[0];
  const int*   ids  = (const int*)d_in[1];
  float* ws  = (float*)d_ws;
  float* out = (float*)d_out;
  const int nrows = in_sizes[1];  // 262144

  cd_zero<<<(WS_TOTAL + 255) / 256, 256, 0, stream>>>(ws);
  // 2048 blocks x 8 waves = 16384 waves, 16 rows/wave grid-stride.
  cd_accum<<<2048, 256, 0, stream>>>(feat, ids, ws, nrows);
  cd_finalize<<<(NUM_IDS + 15) / 16, 32, 0, stream>>>(ws);
  cd_final<<<1, 1, 0, stream>>>(ws, out);
}